// BAttentionTop_26560077758733
// MI455X (gfx1250) — compile-verified
//
#include <hip/hip_runtime.h>
#include <hip/hip_bf16.h>
#include <math.h>
#include <stdint.h>

typedef __attribute__((ext_vector_type(2))) float v2f;
typedef __attribute__((ext_vector_type(8))) float v8f;

#define S_LEN 8192
#define D_DIM 256
#define B_N   32
#define ROWS_PER_WG 128     // 8 waves * 16 rows each
#define CHUNK_K 64          // K columns staged per async chunk
#define NCHUNK  (D_DIM / CHUNK_K)
#define LDS_STRIDE 68       // 64 + 4 pad -> conflict-free (68 mod 64 == 4)
#define HOT_MAX 1024

// builtin signature (from clang diagnostic): param0 = int4 addrspace(1)*, non-const
typedef int b128_t __attribute__((vector_size(16)));
typedef __attribute__((address_space(1))) b128_t gbl_b128;
typedef __attribute__((address_space(3))) b128_t lds_b128;

// ---- CDNA5 async copy: global -> LDS (ASYNCcnt-tracked) ----
__device__ __forceinline__ void async_b128(const float* g, void* l) {
#if __has_builtin(__builtin_amdgcn_global_load_async_to_lds_b128)
    __builtin_amdgcn_global_load_async_to_lds_b128(
        (gbl_b128*)g, (lds_b128*)l, 0, 0);
#else
    uint32_t loff = (uint32_t)(uintptr_t)(lds_b128*)l;
    asm volatile("global_load_async_to_lds_b128 %0, %1, off"
                 :: "v"(loff), "v"(g) : "memory");
#endif
}

__device__ __forceinline__ void wait_async8() {
#if __has_builtin(__builtin_amdgcn_s_wait_asynccnt)
    __builtin_amdgcn_s_wait_asynccnt(8);
#else
    asm volatile("s_wait_asynccnt 0x8" ::: "memory");
#endif
}
__device__ __forceinline__ void wait_async0() {
#if __has_builtin(__builtin_amdgcn_s_wait_asynccnt)
    __builtin_amdgcn_s_wait_asynccnt(0);
#else
    asm volatile("s_wait_asynccnt 0x0" ::: "memory");
#endif
}

// stage K-chunk c (128 rows x 64 cols) into LDS buffer: 8 async b128 per thread
__device__ __forceinline__ void issue_chunk(const float* gx, float* lbuf, int c, int t) {
    const float* gc = gx + c * CHUNK_K;
    #pragma unroll
    for (int i = 0; i < 8; ++i) {
        int idx = t + i * 256;
        int row = idx >> 4;                 // 16 b128 packets per row
        int c4  = (idx & 15) << 2;          // float offset inside chunk row
        async_b128(gc + (long)row * D_DIM + c4, lbuf + row * LDS_STRIDE + c4);
    }
}

// ---------------- init: zero the colsum accumulators ----------------
__global__ void zero_ws_kernel(float* __restrict__ colsum) {
    int i = blockIdx.x * blockDim.x + threadIdx.x;
    if (i < B_N * D_DIM) colsum[i] = 0.0f;
}

// ---- pass A: et = tanh(x·W) via WMMA f32 16x16x4 fed from async-LDS tiles, plus colsum ----
__global__ __launch_bounds__(256)
void et_colsum_kernel(const float* __restrict__ x, const float* __restrict__ W,
                      float* __restrict__ et, float* __restrict__ colsum) {
    const int b       = blockIdx.y;
    const int chunkWG = blockIdx.x;
    const int t       = threadIdx.x;
    const int wave    = t >> 5;
    const int lane    = t & 31;

    __shared__ float wlds[D_DIM];
    __shared__ float tile[2][ROWS_PER_WG * LDS_STRIDE];

    if (t < D_DIM) wlds[t] = W[t];

    const float* gx = x + ((long)b * S_LEN + (long)chunkWG * ROWS_PER_WG) * D_DIM;

    // prologue: two chunks in flight
    issue_chunk(gx, tile[0], 0, t);
    issue_chunk(gx, tile[1], 1, t);

    const int off  = (lane >> 4) << 1;          // K sub-offset 0 or 2 (A/B f32 WMMA layout)
    const int rowl = wave * 16 + (lane & 15);   // this lane's row inside the tile
    v8f acc = {};

    for (int c = 0; c < NCHUNK; ++c) {
        if (c < NCHUNK - 1) wait_async8(); else wait_async0();
        __syncthreads();                         // chunk c visible to all waves

        const float* buf  = tile[c & 1];
        const float* arow = buf + rowl * LDS_STRIDE + off;
        const float* brow = wlds + c * CHUNK_K + off;
        #pragma unroll 4
        for (int k0 = 0; k0 < CHUNK_K; k0 += 4) {
            v2f a  = *(const v2f*)(arow + k0);   // A 16x4 f32 tile feed (conflict-free)
            v2f bb = *(const v2f*)(brow + k0);   // W chunk broadcast into all 16 B columns
            acc = __builtin_amdgcn_wmma_f32_16x16x4_f32(false, a, false, bb,
                                                        (short)0, acc, false, false);
        }

        // per-chunk column sums from the LDS copy (single global read of x overall)
        const int colL = t & 63;
        const int r0   = (t >> 6) * 32;
        float cs = 0.0f;
        #pragma unroll 8
        for (int r = 0; r < 32; ++r) cs += buf[(r0 + r) * LDS_STRIDE + colL];
        atomicAdd(&colsum[b * D_DIM + c * CHUNK_K + colL], cs);

        __syncthreads();                         // all waves done reading buffer c&1
        if (c + 2 < NCHUNK) issue_chunk(gx, (float*)tile[c & 1], c + 2, t);
    }

    // D layout: VGPR v, lanes 0-15 -> M=v, lanes 16-31 -> M=v+8; column 0 = dot(row, W)
    if ((lane & 15) == 0) {
        int base = b * S_LEN + chunkWG * ROWS_PER_WG + wave * 16 + ((lane >> 4) << 3);
        #pragma unroll
        for (int v = 0; v < 8; ++v) et[base + v] = tanhf(acc[v]);
    }
}

// ---------------- pass B helpers ----------------
__device__ __forceinline__ void ins5(float* tp, float v) {
    if (v > tp[4]) {
        tp[4] = v;
        #pragma unroll
        for (int j = 4; j > 0; --j)
            if (tp[j] > tp[j - 1]) { float tmp = tp[j - 1]; tp[j - 1] = tp[j]; tp[j] = tmp; }
    }
}

// ---------------- pass B: top-5 threshold, softmax constants, sparse gather ----------------
__global__ __launch_bounds__(256)
void finalize_kernel(const float* __restrict__ x, const float* __restrict__ et,
                     const float* __restrict__ colsum, float* __restrict__ out) {
    const int b = blockIdx.x;
    const int t = threadIdx.x;
    const float NEG = -1e30f;

    __shared__ float cand[256 * 5];
    __shared__ float mtop[16 * 5];
    __shared__ float shv[4];        // thr, m, invZ, c
    __shared__ float redF[256];
    __shared__ int   redI[256];
    __shared__ int   scan[256];
    __shared__ int   hotS[HOT_MAX];
    __shared__ float hotW[HOT_MAX];

    const float* etb = et + b * S_LEN;

    float tp[5] = {NEG, NEG, NEG, NEG, NEG};
    for (int i = 0; i < S_LEN / 256; ++i) ins5(tp, etb[t + i * 256]);
    #pragma unroll
    for (int j = 0; j < 5; ++j) cand[t * 5 + j] = tp[j];
    __syncthreads();

    if (t < 16) {
        float tq[5] = {NEG, NEG, NEG, NEG, NEG};
        for (int i = t * 80; i < t * 80 + 80; ++i) ins5(tq, cand[i]);
        #pragma unroll
        for (int j = 0; j < 5; ++j) mtop[t * 5 + j] = tq[j];
    }
    __syncthreads();
    if (t == 0) {
        float tq[5] = {NEG, NEG, NEG, NEG, NEG};
        for (int i = 0; i < 80; ++i) ins5(tq, mtop[i]);
        shv[0] = tq[4];                 // thr = 5th largest
        shv[1] = fmaxf(tq[0], 0.0f);    // m
    }
    __syncthreads();
    const float thr = shv[0], m = shv[1];

    int cnt = 0; float se = 0.0f;
    for (int i = 0; i < S_LEN / 256; ++i) {
        float v = etb[t + i * 256];
        if (v >= thr) { ++cnt; se += __expf(v - m); }
    }
    redF[t] = se; redI[t] = cnt;
    __syncthreads();
    if (t == 0) {
        float S = 0.0f; int C = 0;
        for (int i = 0; i < 256; ++i) { S += redF[i]; C += redI[i]; }
        float Z = S + (float)(S_LEN - C) * __expf(-m);
        shv[2] = 1.0f / Z;
        shv[3] = __expf(-m) / Z;
    }

    scan[t] = cnt;
    __syncthreads();
    for (int o = 1; o < 256; o <<= 1) {
        int v = (t >= o) ? scan[t - o] : 0;
        __syncthreads();
        scan[t] += v;
        __syncthreads();
    }
    const float invZ = shv[2], c = shv[3];
    int pos = scan[t] - cnt;
    int total = scan[255];
    for (int i = 0; i < S_LEN / 256; ++i) {
        int s = t + i * 256;
        float v = etb[s];
        if (v >= thr) {
            if (pos < HOT_MAX) { hotS[pos] = s; hotW[pos] = __expf(v - m) * invZ - c; }
            ++pos;
        }
    }
    __syncthreads();
    if (total > HOT_MAX) total = HOT_MAX;

    float accd = c * colsum[b * D_DIM + t];
    const float* xb = x + (long)b * S_LEN * D_DIM;
    for (int i = 0; i < total; ++i)
        accd += hotW[i] * xb[(long)hotS[i] * D_DIM + t];
    out[b * D_DIM + t] = accd;
}

// ---------------- launcher ----------------
extern "C" void kernel_launch(void* const* d_in, const int* in_sizes, int n_in,
                              void* d_out, int out_size, void* d_ws, size_t ws_size,
                              hipStream_t stream) {
    const float* x = (const float*)d_in[0];   // (32, 8192, 256) f32
    const float* W = (const float*)d_in[1];   // (256, 1) f32
    float* out = (float*)d_out;               // (32, 256) f32

    float* ws     = (float*)d_ws;
    float* et     = ws;                       // 32*8192 floats
    float* colsum = ws + (size_t)B_N * S_LEN; // 32*256 floats

    zero_ws_kernel<<<dim3((B_N * D_DIM + 255) / 256), dim3(256), 0, stream>>>(colsum);
    et_colsum_kernel<<<dim3(S_LEN / ROWS_PER_WG, B_N), dim3(256), 0, stream>>>(x, W, et, colsum);
    finalize_kernel<<<dim3(B_N), dim3(256), 0, stream>>>(x, et, colsum, out);
}